// AttentionLayer_16381005267113
// MI455X (gfx1250) — compile-verified
//
#include <hip/hip_runtime.h>
#include <math.h>

// ---------------------------------------------------------------------------
// Additive attention (BiDAF-style) for MI455X / gfx1250.
// B=4, CL=1024, QL=64, H=256, 2H=512.
//   cp = context @ Wc          -> WMMA f32 16x16x4 GEMM     [4096,512]x[512,256]
//   qp = question @ Wq         -> WMMA f32 16x16x4 GEMM     [ 256,512]x[512,256]
//   scores = v . tanh(cp+qp)   -> v_tanh_f32 (TRANS) over TDM-staged LDS tiles
//   softmax over QL=64
//   attended = w @ question    -> WMMA f32 16x16x4 GEMM     [16,64]x[64,512]
//   out = concat(context, attended), plus attention weights
// LDS staging of qp/cp uses the Tensor Data Mover (tensor_load_to_lds) with
// pad_enable to realize the bank-conflict-free 257-float row stride in HW.
// ---------------------------------------------------------------------------

typedef float v2f __attribute__((ext_vector_type(2)));
typedef float v8f __attribute__((ext_vector_type(8)));
typedef unsigned int u32;
typedef unsigned long long u64;
typedef u32 u32x4 __attribute__((ext_vector_type(4)));
typedef u32 u32x8 __attribute__((ext_vector_type(8)));

#define B_   4
#define CL_  1024
#define QL_  64
#define H_   256
#define D2_  512   // 2*H

__device__ __forceinline__ float fast_tanh(float x) {
#if __has_builtin(__builtin_amdgcn_tanhf)
  return __builtin_amdgcn_tanhf(x);   // v_tanh_f32 on gfx1250
#else
  return tanhf(x);
#endif
}

__device__ __forceinline__ void wait_tensorcnt0() {
#if __has_builtin(__builtin_amdgcn_s_wait_tensorcnt)
  __builtin_amdgcn_s_wait_tensorcnt(0);
#else
  asm volatile("s_wait_tensorcnt 0x0" ::: "memory");
#endif
}

// Issue one TDM load: copy `rows` x 256 f32 from `gsrc` (row-major, contiguous)
// into LDS at byte offset `lds_off`, inserting 1 DWORD of padding after every
// 256 DWORDs (-> 257-float LDS row stride). D# packing per CDNA5 ISA §8.3/8.4.
__device__ __forceinline__ void tdm_load_rows(const float* gsrc, u32 lds_off, u32 rows) {
  const u64 ga = (u64)gsrc;
  u32x4 g0;
  g0.x = 1u;                                        // count=1 (valid), no gather
  g0.y = lds_off;                                   // lds_addr (bytes)
  g0.z = (u32)ga;                                   // global_addr[31:0]
  g0.w = (u32)((ga >> 32) & 0x1FFFFFFu) | (2u << 30); // addr[56:32] | type=2
  u32x8 g1;
  g1.s0 = (2u << 16)      // data_size = 2 (4 bytes)
        | (1u << 20)      // pad_enable
        | (7u << 22)      // pad_interval = 256 DWORDs
        | (0u << 25);     // pad_amount   = 1 DWORD
  g1.s1 = (u32)(H_ & 0xFFFF) << 16;                 // tensor_dim0[15:0] = 256
  g1.s2 = (rows & 0xFFFF) << 16;                    // dim0[31:16]=0 | tensor_dim1=rows
  g1.s3 = (u32)(H_ & 0xFFFF) << 16;                 // dim1[31:16]=0 | tile_dim0 = 256
  g1.s4 = rows & 0xFFFF;                            // tile_dim1 = rows, tile_dim2 = 0
  g1.s5 = (u32)H_;                                  // tensor_dim0_stride = 256
  g1.s6 = (u32)((rows * H_) & 0xFFFF) << 16;        // d0s[47:32]=0 | d1stride[15:0]
  g1.s7 = (u32)((rows * H_) >> 16);                 // d1stride[47:16]
  asm volatile("tensor_load_to_lds %0, %1" :: "s"(g0), "s"(g1) : "memory");
}

// C[M,256] = A[M,512] @ W[512,256]; one 16x16 output tile per wave.
// V_WMMA_F32_16X16X4_F32 operand layout (ISA 7.12.2, 32-bit A 16x4):
//   lane&15 -> M row (A) / N col (B,C);  lane>>4 selects K pair {0,1} vs {2,3}.
__global__ __launch_bounds__(128) void proj_wmma(const float* __restrict__ A,
                                                 const float* __restrict__ W,
                                                 float* __restrict__ C) {
  const int lane = threadIdx.x;            // 0..31
  const int wv   = threadIdx.y;            // 0..3
  const int mt   = blockIdx.x;             // M/16 tiles
  const int nt   = blockIdx.y * 4 + wv;    // N/16 tiles (N=256 -> 16 tiles)
  const int m15  = lane & 15;
  const int half = lane >> 4;

  const float* Arow = A + (mt * 16 + m15) * D2_;
  const int    ncol = nt * 16 + m15;

  v8f acc = {};
  for (int k0 = 0; k0 < D2_; k0 += 4) {
    const int kk = k0 + 2 * half;
    v2f a = *(const v2f*)(Arow + kk);            // A[m][kk], A[m][kk+1] (8B aligned)
    v2f b;
    b.x = W[kk * H_ + ncol];                     // B[kk  ][n]
    b.y = W[(kk + 1) * H_ + ncol];               // B[kk+1][n]
    acc = __builtin_amdgcn_wmma_f32_16x16x4_f32(false, a, false, b,
                                                (short)0, acc, false, false);
  }
  // C/D layout: VGPR j, lane -> row = half*8 + j, col = lane&15
  float* Cout = C + (mt * 16 + half * 8) * H_ + ncol;
#pragma unroll
  for (int j = 0; j < 8; ++j) Cout[j * H_] = acc[j];
}

// Fused: scores -> softmax -> attended (WMMA) -> concat. 16 context rows/block.
__global__ __launch_bounds__(256) void attn_fused(
    const float* __restrict__ context,   // [B,CL,512]
    const float* __restrict__ question,  // [B,QL,512]
    const int*   __restrict__ mask,      // [B,CL]
    const float* __restrict__ cp,        // [B,CL,256]
    const float* __restrict__ qp,        // [B,QL,256]
    const float* __restrict__ v,         // [256]
    float* __restrict__ out_ctx,         // [B,CL,1024]
    float* __restrict__ out_w)           // [B,CL,64]
{
  __shared__ float qp_s[QL_][H_ + 1];    // 64 x 257 (TDM pad -> conflict-free)
  __shared__ float cp_s[16][H_ + 1];     // 16 x 257
  __shared__ float w_s[16][QL_ + 1];     // 16 x 65
  __shared__ float v_s[H_];

  const int tid = threadIdx.y * 32 + threadIdx.x; // 0..255
  const int b   = blockIdx.y;
  const int c0  = blockIdx.x * 16;

  // ---- TDM: DMA qp[b] (64x256) and cp rows (16x256) into padded LDS tiles ----
  const float* qpb = qp + b * QL_ * H_;
  const float* cpb = cp + (b * CL_ + c0) * H_;
  if (threadIdx.y == 0) {
    tdm_load_rows(qpb, (u32)(u64)(const void*)&qp_s[0][0], QL_);
    wait_tensorcnt0();
  } else if (threadIdx.y == 1) {
    tdm_load_rows(cpb, (u32)(u64)(const void*)&cp_s[0][0], 16);
    wait_tensorcnt0();
  }
  v_s[tid] = v[tid];
  __syncthreads();   // publishes TDM-written LDS to all 8 waves

  // ---- scores: each thread owns 4 (row,q) pairs; 256-deep tanh reduction ----
#pragma unroll
  for (int i = 0; i < 4; ++i) {
    const int p = i * 256 + tid;
    const int r = p >> 6, q = p & 63;
    float s = 0.f;
#pragma unroll 8
    for (int h = 0; h < H_; ++h)
      s = fmaf(v_s[h], fast_tanh(cp_s[r][h] + qp_s[q][h]), s);
    s = fminf(fmaxf(s, -10.f), 10.f);
    if (mask[b * CL_ + c0 + r] == 0) s = -10.f;
    w_s[r][q] = s;
  }
  __syncthreads();

  // ---- softmax over q (64) per row; 16 rows on threads 0..15 ----
  if (tid < 16) {
    float m = -1e30f;
    for (int q = 0; q < QL_; ++q) m = fmaxf(m, w_s[tid][q]);
    float sum = 0.f;
    for (int q = 0; q < QL_; ++q) {
      float e = __expf(w_s[tid][q] - m);
      w_s[tid][q] = e;
      sum += e;
    }
    const float inv = 1.f / sum;
    float* wo = out_w + (b * CL_ + c0 + tid) * QL_;
    for (int q = 0; q < QL_; ++q) {
      float wq = w_s[tid][q] * inv;
      w_s[tid][q] = wq;
      wo[q] = wq;
    }
  }
  __syncthreads();

  // ---- attended = w[16x64] @ question[b][64x512] via WMMA f32 16x16x4 ----
  const int lane = threadIdx.x, wv = threadIdx.y;   // 8 waves, 4 N-tiles each
  const int m15 = lane & 15, half = lane >> 4;
  const float* Qb = question + b * QL_ * D2_;
#pragma unroll
  for (int it = 0; it < 4; ++it) {
    const int d0 = (wv * 4 + it) * 16;              // 32 tiles cover d=0..511
    v8f acc = {};
#pragma unroll
    for (int k0 = 0; k0 < QL_; k0 += 4) {
      const int kk = k0 + 2 * half;
      v2f a  = { w_s[m15][kk], w_s[m15][kk + 1] };
      v2f bb = { Qb[kk * D2_ + d0 + m15], Qb[(kk + 1) * D2_ + d0 + m15] };
      acc = __builtin_amdgcn_wmma_f32_16x16x4_f32(false, a, false, bb,
                                                  (short)0, acc, false, false);
    }
#pragma unroll
    for (int j = 0; j < 8; ++j)
      out_ctx[(b * CL_ + c0 + half * 8 + j) * 1024 + 512 + d0 + m15] = acc[j];
  }

  // ---- concat: copy context rows into out[:, 0:512] (float4, coalesced) ----
  const float4* src = (const float4*)(context + (b * CL_ + c0) * D2_);
#pragma unroll
  for (int i = 0; i < 8; ++i) {
    const int idx = i * 256 + tid;        // 2048 float4s = 16 rows x 128
    const int r = idx >> 7, pos = idx & 127;
    const float4 val = src[idx];
    *(float4*)(out_ctx + (b * CL_ + c0 + r) * 1024 + pos * 4) = val;
  }
}

extern "C" void kernel_launch(void* const* d_in, const int* in_sizes, int n_in,
                              void* d_out, int out_size, void* d_ws, size_t ws_size,
                              hipStream_t stream) {
  const float* context  = (const float*)d_in[0];  // [4,1024,512]
  const float* question = (const float*)d_in[1];  // [4,64,512]
  const int*   mask     = (const int*)d_in[2];    // [4,1024]
  const float* Wc       = (const float*)d_in[3];  // [512,256]
  const float* Wq       = (const float*)d_in[4];  // [512,256]
  const float* v        = (const float*)d_in[5];  // [256]

  float* cp = (float*)d_ws;                       // [4096,256] = 4 MB
  float* qp = cp + B_ * CL_ * H_;                 // [256,256]  = 256 KB
  float* out_ctx = (float*)d_out;                 // [4,1024,1024]
  float* out_w   = out_ctx + B_ * CL_ * 1024;     // [4,1024,64]

  // cp = context @ Wc : M=4096 -> 256 M-tiles; 16 N-tiles as grid.y(4) x waves(4)
  proj_wmma<<<dim3((B_ * CL_) / 16, 4), dim3(32, 4), 0, stream>>>(context, Wc, cp);
  // qp = question @ Wq : M=256 -> 16 M-tiles
  proj_wmma<<<dim3((B_ * QL_) / 16, 4), dim3(32, 4), 0, stream>>>(question, Wq, qp);
  // fused scores/softmax/attended/concat: 16 context rows per block
  attn_fused<<<dim3(CL_ / 16, B_), dim3(32, 8), 0, stream>>>(
      context, question, mask, cp, qp, v, out_ctx, out_w);
}